// Local_rep_layer_10995116277869
// MI455X (gfx1250) — compile-verified
//
#include <hip/hip_runtime.h>
#include <hip/hip_bf16.h>
#include <math.h>

// ---------------------------------------------------------------------------
// Local_rep_layer on MI455X (gfx1250, wave32, WMMA)
// B=8, E=64, M=128, S=32, D=256, H=8, dk=32
// ---------------------------------------------------------------------------

typedef __attribute__((ext_vector_type(2))) float v2f;
typedef __attribute__((ext_vector_type(8))) float v8f;

static __device__ __forceinline__ v8f wmma_f32_k4(v2f a, v2f b, v8f c) {
  // V_WMMA_F32_16X16X4_F32: D = A(16x4) * B(4x16) + C(16x16), full f32.
  return __builtin_amdgcn_wmma_f32_16x16x4_f32(
      /*neg_a=*/false, a, /*neg_b=*/false, b,
      /*c_mod=*/(short)0, c, /*reuse_a=*/false, /*reuse_b=*/false);
}

// ---------------------------------------------------------------------------
// Kernel 1: row-gathered GEMM  Y[rows,256] = X[rows,256] @ W[256,256] + bias
//   use_gather=1: X row r -> sentences[(r>>7)*32 + sent_ids[r]] (k projection)
//   one wave per 16x16 output tile, 8 waves per block, grid = rows/8 blocks
// ---------------------------------------------------------------------------
__global__ __launch_bounds__(256) void proj_gemm(
    const float* __restrict__ X, const int* __restrict__ sids,
    const float* __restrict__ gsrc, int use_gather,
    const float* __restrict__ W, const float* __restrict__ bias,
    float* __restrict__ Y) {
  const int tid  = threadIdx.x;
  const int wid  = tid >> 5;
  const int lane = tid & 31;
  const int hi   = (lane >> 4);   // 0: K pair {0,1}; 1: K pair {2,3}
  const int ln   = lane & 15;     // row-in-tile (A), col-in-tile (B/C/D)

  const int tile = blockIdx.x * 8 + wid;
  const int mt = tile >> 4;       // 16 column tiles (N=256)
  const int nt = tile & 15;

  const int row = mt * 16 + ln;   // global A row for this lane
  const float* xr;
  if (use_gather) {
    const int bb = row >> 7;                 // batch (M=128 rows per batch)
    const int m  = row & 127;
    const int sid = sids[bb * 128 + m];
    xr = gsrc + (size_t)(bb * 32 + sid) * 256;  // sentences[b, sid, :]
  } else {
    xr = X + (size_t)row * 256;
  }

  v8f c = {};
  const int coln = nt * 16 + ln;
  #pragma unroll 8
  for (int k0 = 0; k0 < 256; k0 += 4) {
    const int ka = k0 + 2 * hi;
    v2f a, bm;
    a.x  = xr[ka];
    a.y  = xr[ka + 1];
    bm.x = W[(size_t)ka * 256 + coln];
    bm.y = W[(size_t)(ka + 1) * 256 + coln];
    c = wmma_f32_k4(a, bm, c);
  }

  const float bn = bias[coln];
  #pragma unroll
  for (int r = 0; r < 8; ++r) {
    const int mrow = mt * 16 + r + 8 * hi;   // C/D layout: VGPR r -> M=r / M=8+r
    Y[(size_t)mrow * 256 + coln] = c[r] + bn;
  }
}

// ---------------------------------------------------------------------------
// Kernel 2: fused grouped attention + output projection.
// Block (g, b): mask entity g in batch b.
//   Phase A: compact {m : eid[b,m]==g}   (serial on thread 0 -> deterministic)
//   Phase B: for each (query entity qe, head h): softmax over group, weighted
//            sum of v -> ctx[qe, h*32+d] in LDS (row stride 258: no bank conf)
//   Phase C: out[64,256] = ctx @ W[3] + b[3] via f32 WMMA, direct to d_out.
//            transposed=0 -> rep_h rows g*64+qe ; =1 -> rep_t rows qe*64+g
// ---------------------------------------------------------------------------
#define CTX_LD 258

__global__ __launch_bounds__(256) void attn_fused(
    const float* __restrict__ qp, const float* __restrict__ kp,
    const float* __restrict__ vp, const int* __restrict__ eid,
    const float* __restrict__ W3, const float* __restrict__ b3,
    float* __restrict__ out, int transposed) {
  __shared__ float ctx[64 * CTX_LD];
  __shared__ int   list[128];
  __shared__ int   cnt_s;
  __shared__ float meanv[256];

  const int g = blockIdx.x;   // mask entity
  const int b = blockIdx.y;   // batch
  const int tid = threadIdx.x;

  if (tid == 0) {
    int c = 0;
    for (int m = 0; m < 128; ++m)
      if (eid[b * 128 + m] == g) list[c++] = m;
    cnt_s = c;
  }
  __syncthreads();
  const int cnt = cnt_s;

  if (cnt == 0) {
    // all-masked: softmax over 128 equal (-1e9) scores -> uniform mean of v
    float s = 0.f;
    for (int m = 0; m < 128; ++m)
      s += vp[(size_t)(b * 128 + m) * 256 + tid];
    meanv[tid] = s * (1.0f / 128.0f);
  }
  __syncthreads();

  const float scale = 0.17677669529663687f;  // 1/sqrt(32)

  for (int p = tid; p < 512; p += 256) {     // 64 query entities x 8 heads
    const int qe = p >> 3;
    const int h  = p & 7;
    float* crow = &ctx[qe * CTX_LD + h * 32];
    if (cnt == 0) {
      #pragma unroll
      for (int d = 0; d < 32; ++d) crow[d] = meanv[h * 32 + d];
    } else {
      const float* qv = qp + (size_t)(b * 64 + qe) * 256 + h * 32;
      float qr[32];
      #pragma unroll
      for (int d = 0; d < 32; ++d) qr[d] = qv[d];

      float mx = -3.0e38f;
      for (int t = 0; t < cnt; ++t) {
        const float* kv = kp + (size_t)(b * 128 + list[t]) * 256 + h * 32;
        float s = 0.f;
        #pragma unroll
        for (int d = 0; d < 32; ++d) s += qr[d] * kv[d];
        s *= scale;
        mx = fmaxf(mx, s);
      }

      float acc[32];
      #pragma unroll
      for (int d = 0; d < 32; ++d) acc[d] = 0.f;
      float wsum = 0.f;
      for (int t = 0; t < cnt; ++t) {
        const int m = list[t];
        const float* kv = kp + (size_t)(b * 128 + m) * 256 + h * 32;
        float s = 0.f;
        #pragma unroll
        for (int d = 0; d < 32; ++d) s += qr[d] * kv[d];
        const float w = expf(s * scale - mx);
        wsum += w;
        const float* vv = vp + (size_t)(b * 128 + m) * 256 + h * 32;
        #pragma unroll
        for (int d = 0; d < 32; ++d) acc[d] += w * vv[d];
      }
      const float inv = 1.0f / wsum;
      #pragma unroll
      for (int d = 0; d < 32; ++d) crow[d] = acc[d] * inv;
    }
  }
  __syncthreads();

  // ---- Phase C: out = ctx @ W3 + b3 (64x256 @ 256x256), f32 WMMA ----
  const int wid  = tid >> 5;
  const int lane = tid & 31;
  const int hi   = (lane >> 4);
  const int ln   = lane & 15;

  #pragma unroll
  for (int tt = 0; tt < 8; ++tt) {
    const int t  = wid * 8 + tt;   // 64 tiles: 4 row-tiles x 16 col-tiles
    const int mt = t >> 4;
    const int nt = t & 15;
    const int coln = nt * 16 + ln;

    const float* arow = &ctx[(mt * 16 + ln) * CTX_LD];
    v8f c = {};
    #pragma unroll 8
    for (int k0 = 0; k0 < 256; k0 += 4) {
      const int ka = k0 + 2 * hi;
      v2f a, bm;
      a.x  = arow[ka];
      a.y  = arow[ka + 1];
      bm.x = W3[(size_t)ka * 256 + coln];
      bm.y = W3[(size_t)(ka + 1) * 256 + coln];
      c = wmma_f32_k4(a, bm, c);
    }

    const float bn = b3[coln];
    #pragma unroll
    for (int r = 0; r < 8; ++r) {
      const int mrow = mt * 16 + r + 8 * hi;  // query-entity row 0..63
      const int rowg = transposed ? (b * 4096 + mrow * 64 + g)
                                  : (b * 4096 + g * 64 + mrow);
      out[(size_t)rowg * 256 + coln] = c[r] + bn;
    }
  }
}

// ---------------------------------------------------------------------------
extern "C" void kernel_launch(void* const* d_in, const int* in_sizes, int n_in,
                              void* d_out, int out_size, void* d_ws, size_t ws_size,
                              hipStream_t stream) {
  const float* entities  = (const float*)d_in[0];  // [8,64,256]
  const float* mentions  = (const float*)d_in[1];  // [8,128,256]
  const float* sentences = (const float*)d_in[2];  // [8,32,256]
  const int*   sent_ids  = (const int*)d_in[3];    // [8,128]
  const int*   ment_eid  = (const int*)d_in[4];    // [8,128]
  const float* W_h = (const float*)d_in[5];        // [4,256,256]
  const float* b_h = (const float*)d_in[6];        // [4,256]
  const float* W_t = (const float*)d_in[7];
  const float* b_t = (const float*)d_in[8];
  float* outp = (float*)d_out;

  // workspace layout (floats): qh qt | kh kt vh vt  (~5 MB total)
  float* qh = (float*)d_ws;
  float* qt = qh + 8 * 64 * 256;
  float* kh = qt + 8 * 64 * 256;
  float* kt = kh + 8 * 128 * 256;
  float* vh = kt + 8 * 128 * 256;
  float* vt = vh + 8 * 128 * 256;

  const size_t WS = 256 * 256;  // one [D,D] weight slab

  // q projections: entities @ W[0]  (512 rows -> 64 blocks)
  proj_gemm<<<64, 256, 0, stream>>>(entities, nullptr, nullptr, 0,
                                    W_h + 0 * WS, b_h + 0 * 256, qh);
  proj_gemm<<<64, 256, 0, stream>>>(entities, nullptr, nullptr, 0,
                                    W_t + 0 * WS, b_t + 0 * 256, qt);
  // k projections: gathered sentences @ W[1]  (1024 rows -> 128 blocks)
  proj_gemm<<<128, 256, 0, stream>>>(nullptr, sent_ids, sentences, 1,
                                     W_h + 1 * WS, b_h + 1 * 256, kh);
  proj_gemm<<<128, 256, 0, stream>>>(nullptr, sent_ids, sentences, 1,
                                     W_t + 1 * WS, b_t + 1 * 256, kt);
  // v projections: mentions @ W[2]
  proj_gemm<<<128, 256, 0, stream>>>(mentions, nullptr, nullptr, 0,
                                     W_h + 2 * WS, b_h + 2 * 256, vh);
  proj_gemm<<<128, 256, 0, stream>>>(mentions, nullptr, nullptr, 0,
                                     W_t + 2 * WS, b_t + 2 * 256, vt);

  // fused grouped attention + output projection
  dim3 grid(64, 8);
  float* rep_h = outp;
  float* rep_t = outp + (size_t)8 * 4096 * 256;
  attn_fused<<<grid, 256, 0, stream>>>(qh, kh, vh, ment_eid,
                                       W_h + 3 * WS, b_h + 3 * 256, rep_h, 0);
  attn_fused<<<grid, 256, 0, stream>>>(qt, kt, vt, ment_eid,
                                       W_t + 3 * WS, b_t + 3 * 256, rep_t, 1);
}